// PairwiseHead_76416058131342
// MI455X (gfx1250) — compile-verified
//
#include <hip/hip_runtime.h>

#define NROWS 8192
#define FDIM  128
#define TILE  64
#define LSTR  132          // padded LDS row stride in floats (132%64=4 -> conflict-free, 16B-aligned rows)

typedef __attribute__((ext_vector_type(2))) float v2f;
typedef __attribute__((ext_vector_type(8))) float v8f;

// ---------------------------------------------------------------------------
// Kernel 1: per-row squared norms + per-block sum(X^2) partials.
// One wave (32 lanes) per row; lane loads 4 contiguous floats.
// ---------------------------------------------------------------------------
__global__ __launch_bounds__(256) void rowsq_kernel(const float* __restrict__ X,
                                                    float* __restrict__ sq,
                                                    float* __restrict__ sumsq_part) {
    __shared__ float wsum[8];
    const int wave = threadIdx.x >> 5;
    const int lane = threadIdx.x & 31;
    const int row  = blockIdx.x * 8 + wave;
    const float4 v = ((const float4*)(X + row * FDIM))[lane];
    float s = v.x * v.x + v.y * v.y + v.z * v.z + v.w * v.w;
    #pragma unroll
    for (int off = 16; off > 0; off >>= 1) s += __shfl_xor(s, off, 32);
    if (lane == 0) { sq[row] = s; wsum[wave] = s; }
    __syncthreads();
    if (threadIdx.x == 0) {
        float t = 0.f;
        #pragma unroll
        for (int i = 0; i < 8; ++i) t += wsum[i];
        sumsq_part[blockIdx.x] = t;
    }
}

// ---------------------------------------------------------------------------
// Kernel 2: fused Gram GEMM (fp32 WMMA 16x16x4) + loss epilogue.
// Block = 128 threads = 4 waves; computes one 64x64 tile of dist.
// Only TJ >= TI tiles are computed (symmetry); off-diagonal weighted x2.
// Per-block partial sums written to fixed slots (deterministic).
// ---------------------------------------------------------------------------
__global__ __launch_bounds__(128) void gram_loss_kernel(const float* __restrict__ X,
                                                        const long long* __restrict__ y,
                                                        const float* __restrict__ sq,
                                                        float* __restrict__ part) {
    const int TI  = blockIdx.y;
    const int TJ  = blockIdx.x;
    const int bid = TI * gridDim.x + TJ;
    if (TI > TJ) {                       // lower triangle: write zeros, exit (uniform branch)
        if (threadIdx.x == 0) { part[2 * bid] = 0.f; part[2 * bid + 1] = 0.f; }
        return;
    }

    __shared__ float As[TILE * LSTR];
    __shared__ float Bs[TILE * LSTR];
    __shared__ float red[8];

    const int tid = threadIdx.x;
    const int I0  = TI * TILE;
    const int J0  = TJ * TILE;

    // Stage both 64x128 tiles with coalesced float4 loads.
    {
        const float4* Xv  = (const float4*)X;
        float4*       As4 = (float4*)As;
        float4*       Bs4 = (float4*)Bs;
        #pragma unroll
        for (int idx = tid; idx < TILE * (FDIM / 4); idx += 128) {
            const int r = idx >> 5;      // row in tile
            const int c = idx & 31;      // float4 column
            As4[r * (LSTR / 4) + c] = Xv[(I0 + r) * (FDIM / 4) + c];
            Bs4[r * (LSTR / 4) + c] = Xv[(J0 + r) * (FDIM / 4) + c];
        }
    }
    __syncthreads();

    const int wave = tid >> 5;
    const int lane = tid & 31;
    const int hl   = lane >> 4;          // lane half (selects K pair / M offset)
    const int l15  = lane & 15;

    v8f acc[4] = {};                     // 4 column sub-tiles of 16x16 f32

    // A fragment base: row = wave*16 + (lane&15), K pair selected by lane half.
    const float* Arow = &As[(wave * 16 + l15) * LSTR + 2 * hl];

    for (int k0 = 0; k0 < FDIM; k0 += 4) {
        const v2f a = *(const v2f*)(Arow + k0);
        #pragma unroll
        for (int n = 0; n < 4; ++n) {
            const v2f b = *(const v2f*)(&Bs[(n * 16 + l15) * LSTR + k0 + 2 * hl]);
            acc[n] = __builtin_amdgcn_wmma_f32_16x16x4_f32(
                /*neg_a=*/false, a, /*neg_b=*/false, b,
                /*c_mod=*/(short)0, acc[n], /*reuse_a=*/false, /*reuse_b=*/false);
        }
    }

    // Epilogue: C/D layout: VGPR r -> M = r + 8*hl, N = lane&15 (per sub-tile n).
    float ll = 0.f, ee = 0.f;
    #pragma unroll
    for (int n = 0; n < 4; ++n) {
        const int       j   = J0 + n * 16 + l15;
        const float     sqj = sq[j];
        const long long yj  = y[j];
        #pragma unroll
        for (int r = 0; r < 8; ++r) {
            const int   i    = I0 + wave * 16 + r + 8 * hl;
            const float g    = acc[n][r];
            const float dist = fmaxf(sq[i] + sqj - 2.0f * g, 0.0f);
            if (y[i] != yj) ll += __logf(1.01f - __expf(-dist));
            else            ee += dist;
        }
    }
    #pragma unroll
    for (int off = 16; off > 0; off >>= 1) {
        ll += __shfl_xor(ll, off, 32);
        ee += __shfl_xor(ee, off, 32);
    }
    if (lane == 0) { red[wave * 2] = ll; red[wave * 2 + 1] = ee; }
    __syncthreads();
    if (tid == 0) {
        const float w = (TI == TJ) ? 1.0f : 2.0f;
        part[2 * bid]     = w * (red[0] + red[2] + red[4] + red[6]);
        part[2 * bid + 1] = w * (red[1] + red[3] + red[5] + red[7]);
    }
}

// ---------------------------------------------------------------------------
// Kernel 3: deterministic final reduction in double precision -> scalar loss.
// ---------------------------------------------------------------------------
__global__ __launch_bounds__(256) void finalize_kernel(const float* __restrict__ sumsq_part,
                                                       const float* __restrict__ part,
                                                       float* __restrict__ out) {
    __shared__ double sl[256], se[256], s2[256];
    const int t = threadIdx.x;
    double a = 0.0, b = 0.0, c = 0.0;
    for (int i = t; i < 16384; i += 256) {
        a += (double)part[2 * i];
        b += (double)part[2 * i + 1];
    }
    for (int i = t; i < 1024; i += 256) c += (double)sumsq_part[i];
    sl[t] = a; se[t] = b; s2[t] = c;
    __syncthreads();
    for (int s = 128; s > 0; s >>= 1) {
        if (t < s) { sl[t] += sl[t + s]; se[t] += se[t + s]; s2[t] += s2[t + s]; }
        __syncthreads();
    }
    if (t == 0) {
        const double invBB = 1.0 / ((double)NROWS * (double)NROWS);
        const double loss  = (-sl[0] + se[0]) * invBB
                           + 0.01 * s2[0] / ((double)NROWS * (double)FDIM);
        out[0] = (float)loss;
    }
}

// ---------------------------------------------------------------------------
extern "C" void kernel_launch(void* const* d_in, const int* in_sizes, int n_in,
                              void* d_out, int out_size, void* d_ws, size_t ws_size,
                              hipStream_t stream) {
    (void)in_sizes; (void)n_in; (void)out_size; (void)ws_size;
    const float*     X = (const float*)d_in[0];
    const long long* y = (const long long*)d_in[1];

    float* ws         = (float*)d_ws;
    float* sq         = ws;                  // 8192 floats
    float* sumsq_part = ws + NROWS;          // 1024 floats
    float* part       = ws + NROWS + 1024;   // 128*128*2 = 32768 floats

    rowsq_kernel<<<NROWS / 8, 256, 0, stream>>>(X, sq, sumsq_part);

    dim3 grid(NROWS / TILE, NROWS / TILE);   // (TJ, TI) = (128, 128)
    gram_loss_kernel<<<grid, 128, 0, stream>>>(X, y, sq, part);

    finalize_kernel<<<1, 256, 0, stream>>>(sumsq_part, part, (float*)d_out);
}